// BlockEquivariantAttentionUpsample_51926154609238
// MI455X (gfx1250) — compile-verified
//
#include <hip/hip_runtime.h>

typedef float v2f __attribute__((ext_vector_type(2)));
typedef float v8f __attribute__((ext_vector_type(8)));

// ---------------- workspace layout (float offsets) ----------------
#define WS_W3J   0          // 13318 floats of Wigner tables
#define WS_SH    13320      // 256 x 6
#define WS_PBV   14856      // 2 x 256
#define WS_ESC   15368      // e4[2], e6[2]
#define WS_ALPHA 15372      // 2 x 8 x 3 folded coefficients
#define WS_FEAT  16384      // 65536 x 22
#define WS_GR    1458176    // 256 blocks x 256 x 24 (normalized, zero-padded)
#define WS_V     3031040    // 256 blocks x 256 x 80 (geometry values, padded)
#define WS_CTX   8273920    // 256 blocks x 256 x 80

// table registry: (l1,l2,l3) and float offset of each W3J table
__device__ const int g_tl1[14]  = {4,4,6,6,4,6,4,4,6,6,4,4,6,6};
__device__ const int g_tl2[14]  = {0,2,2,0,2,2,4,6,4,6,4,6,4,6};
__device__ const int g_tl3[14]  = {4,4,4,6,6,6,4,4,4,4,6,6,6,6};
__device__ const int g_toff[14] = {0,81,486,1071,1240,1825,2670,3399,4452,5505,7026,8079,9600,11121};

struct UE { int col; double re, im; };

__device__ inline int urow(int l, int a, UE* e) {
  const double is2 = 0.70710678118654752440;
  if (a == l) { e[0].col = l; e[0].re = 1.0; e[0].im = 0.0; return 1; }
  if (a > l) {
    int m = a - l; double s = (m & 1) ? -1.0 : 1.0;
    e[0].col = l - m; e[0].re = is2;     e[0].im = 0.0;
    e[1].col = l + m; e[1].re = s * is2; e[1].im = 0.0;
    return 2;
  }
  int m = l - a; double s = (m & 1) ? -1.0 : 1.0;
  e[0].col = l - m; e[0].re = 0.0; e[0].im = is2;
  e[1].col = l + m; e[1].re = 0.0; e[1].im = -s * is2;
  return 2;
}

__device__ inline double cg_coef(const double* F, int l1,int m1,int l2,int m2,int l3,int m3){
  if (m1 + m2 != m3) return 0.0;
  double pre = sqrt((2.0*l3+1.0) * F[l3+l1-l2] * F[l3-l1+l2] * F[l1+l2-l3] / F[l1+l2+l3+1]);
  pre *= sqrt(F[l3+m3]*F[l3-m3]*F[l1-m1]*F[l1+m1]*F[l2-m2]*F[l2+m2]);
  double s = 0.0;
  for (int k = 0; k <= l1+l2-l3; ++k){
    int d2=l1-m1-k, d3=l2+m2-k, d4=l3-l2+m1+k, d5=l3-l1-m2+k;
    if (d2<0||d3<0||d4<0||d5<0) continue;
    double p = F[k]*F[l1+l2-l3-k]*F[d2]*F[d3]*F[d4]*F[d5];
    s += ((k&1)? -1.0 : 1.0) / p;
  }
  return pre * s;
}

// ---------------- init: W3J tables, sh, pbv, folded per-layer scalars ----------
__global__ __launch_bounds__(256) void eqatt_init(
    const float* ls4, const float* ls6, const float* posw, const float* posb,
    const float* wi4, const float* wi6, const float* wv, const float* wo,
    const float* wo4, const float* wo6, float* ws)
{
  __shared__ double F[20];
  int tid = threadIdx.x;
  if (tid < 20) { double r = 1.0; for (int i = 2; i <= tid; ++i) r *= (double)i; F[tid] = r; }
  __syncthreads();

  // W3J tables: real part of U1 x U2 x U3 x w3j_complex (l1+l2+l3 even -> real)
  for (int tab = 0; tab < 14; ++tab) {
    int l1 = g_tl1[tab], l2 = g_tl2[tab], l3 = g_tl3[tab];
    int d2 = 2*l2+1, d3 = 2*l3+1;
    int n = (2*l1+1)*d2*d3;
    for (int e = tid; e < n; e += 256) {
      int a = e/(d2*d3), b = (e/d3)%d2, c = e%d3;
      UE u1[2], u2[2], u3[2];
      int n1 = urow(l1,a,u1), n2 = urow(l2,b,u2), n3 = urow(l3,c,u3);
      double acc = 0.0;
      for (int p=0;p<n1;++p) for (int q=0;q<n2;++q) for (int r=0;r<n3;++r){
        int m1 = u1[p].col-l1, m2 = u2[q].col-l2, m3 = u3[r].col-l3;
        if (m1+m2+m3 != 0) continue;
        int ex = l1 - l2 - m3;
        double sgn = (ex & 1) ? -1.0 : 1.0;
        double wc = sgn / sqrt(2.0*l3+1.0) * cg_coef(F,l1,m1,l2,m2,l3,-m3);
        double re12 = u1[p].re*u2[q].re - u1[p].im*u2[q].im;
        double im12 = u1[p].re*u2[q].im + u1[p].im*u2[q].re;
        double re   = re12*u3[r].re - im12*u3[r].im;
        acc += re * wc;
      }
      ws[g_toff[tab] + e] = (float)acc;
    }
  }

  // spherical-harmonic block table + key-side position bias
  {
    int n = tid;
    int iy = n >> 4, ix = n & 15;
    float y = -1.f + (2.f/15.f)*(float)iy;
    float x = -1.f + (2.f/15.f)*(float)ix;
    float nn = fmaxf(sqrtf(x*x + y*y), 1e-8f);
    x /= nn; y /= nn;
    const float PI = 3.14159265358979323846f;
    float cc = 0.5f*sqrtf(15.f/PI);
    float shv[6];
    shv[0] = 0.5f/sqrtf(PI);
    shv[1] = cc*x*y;
    shv[2] = 0.f;                       // z = 0
    shv[3] = -0.25f*sqrtf(5.f/PI);      // (3z^2-1) scaled
    shv[4] = 0.f;
    shv[5] = 0.5f*cc*(x*x - y*y);
    for (int j=0;j<6;++j) ws[WS_SH + n*6 + j] = shv[j];
    for (int l=0;l<2;++l){
      float pv = posb[l];
      for (int j=0;j<6;++j) pv += shv[j]*posw[l*6+j];
      ws[WS_PBV + l*256 + n] = pv;
    }
  }
  if (tid < 2){ ws[WS_ESC + tid] = __expf(ls4[tid]); ws[WS_ESC + 2 + tid] = __expf(ls6[tid]); }

  // fold wi/wv/wo/wout into 3 scalars per (layer, term)
  if (tid < 16){
    int l = tid >> 3, t = tid & 7;
    float a4 = sqrtf(9.f/24.f),  a6 = sqrtf(13.f/24.f);
    float b4 = sqrtf(9.f/256.f), b6 = sqrtf(13.f/256.f);
    float inv = 1.f/sqrtf(8.f);
    bool srcIs4 = (t==0||t==1||t==4||t==5);   // h4 vs h6 source
    bool useC4  = (t==0||t==2||t==4||t==6);   // c4 vs c6 context
    const float* wis = srcIs4 ? (wi4 + l*8) : (wi6 + l*8);
    const float* wod = (t < 4) ? (wo4 + l*8) : (wo6 + l*8);
    float coeff[8];
    for (int v=0;v<8;++v){
      float cs = 0.f;
      for (int u=0;u<8;++u){
        float wsum = 0.f;
        for (int w=0;w<8;++w) wsum += wo[(((l*8+t)*8+u)*8+v)*8+w]*wod[w];
        cs += wis[u]*wsum;
      }
      coeff[v] = cs;
    }
    float bs = ((t<4)? b4 : b6) * inv * (useC4 ? a4 : a6);
    for (int sp=0; sp<3; ++sp){
      int s = useC4 ? sp : (3+sp);
      const float* wig = (s==2||s==3||s==5) ? (wi6 + l*8) : (wi4 + l*8);
      float acc = 0.f;
      for (int v=0;v<8;++v){
        float g = 0.f;
        for (int u=0;u<8;++u) g += wig[u]*wv[((l*6+s)*8+u)*8+v];
        acc += coeff[v]*g;
      }
      ws[WS_ALPHA + (l*8+t)*3 + sp] = bs*acc;
    }
  }
}

// ---------------- pack f4|f6 -> feat ----------------
__global__ __launch_bounds__(256) void eqatt_pack(const float* f4, const float* f6, float* ws)
{
  int idx = blockIdx.x*256 + threadIdx.x;    // 0..65535
  float* o = ws + WS_FEAT + idx*22;
  const float* a = f4 + idx*9;
  const float* b = f6 + idx*13;
  for (int c=0;c<9;++c)  o[c]   = a[c];
  for (int c=0;c<13;++c) o[9+c] = b[c];
}

// ---------------- per-token: normalized features + geometry value vectors ------
__global__ __launch_bounds__(256) void eqatt_values(float* ws)
{
  __shared__ float sW[2670];                 // W3J tables 0..5
  for (int i = threadIdx.x; i < 2670; i += 256) sW[i] = ws[i];
  __syncthreads();

  int idx = blockIdx.x*256 + threadIdx.x;
  int b = idx >> 14, pix = idx & 16383;
  int row = pix >> 7, col = pix & 127;
  int bb = b*64 + (row>>4)*8 + (col>>4);
  int n  = (row & 15)*16 + (col & 15);

  const float* ft = ws + WS_FEAT + idx*22;
  float f4[9], f6[13];
  float s4 = 0.f, s6 = 0.f;
  for (int i=0;i<9;++i){  f4[i]=ft[i];   s4 += f4[i]*f4[i]; }
  for (int i=0;i<13;++i){ f6[i]=ft[9+i]; s6 += f6[i]*f6[i]; }
  float r4 = 1.f / fmaxf(sqrtf(s4), 1e-12f);
  float r6 = 1.f / fmaxf(sqrtf(s6), 1e-12f);

  float* gr = ws + WS_GR + (bb*256+n)*24;
  for (int i=0;i<9;++i)  gr[i]   = f4[i]*r4;
  for (int i=0;i<13;++i) gr[9+i] = f6[i]*r6;
  gr[22] = 0.f; gr[23] = 0.f;

  float sh0 = ws[WS_SH + n*6];
  float sh2[5];
  for (int j=0;j<5;++j) sh2[j] = ws[WS_SH + n*6 + 1 + j];

  float Vr[66];
  for (int k=0;k<9;++k){  float a=0.f; for (int i=0;i<9;++i)  a += f4[i]*sW[i*9+k];           Vr[k]    = sh0*a; }
  for (int k=0;k<9;++k){  float a=0.f; for (int i=0;i<9;++i){  float fi=f4[i]; for (int j=0;j<5;++j) a += fi*sh2[j]*sW[81  +(i*5+j)*9 +k]; } Vr[9+k]  = a; }
  for (int k=0;k<9;++k){  float a=0.f; for (int i=0;i<13;++i){ float fi=f6[i]; for (int j=0;j<5;++j) a += fi*sh2[j]*sW[486 +(i*5+j)*9 +k]; } Vr[18+k] = a; }
  for (int k=0;k<13;++k){ float a=0.f; for (int i=0;i<13;++i) a += f6[i]*sW[1071+i*13+k];     Vr[27+k] = sh0*a; }
  for (int k=0;k<13;++k){ float a=0.f; for (int i=0;i<9;++i){  float fi=f4[i]; for (int j=0;j<5;++j) a += fi*sh2[j]*sW[1240+(i*5+j)*13+k]; } Vr[40+k] = a; }
  for (int k=0;k<13;++k){ float a=0.f; for (int i=0;i<13;++i){ float fi=f6[i]; for (int j=0;j<5;++j) a += fi*sh2[j]*sW[1825+(i*5+j)*13+k]; } Vr[53+k] = a; }

  float* vp = ws + WS_V + (bb*256+n)*80;
  for (int c=0;c<66;++c)  vp[c] = Vr[c];
  for (int c=66;c<80;++c) vp[c] = 0.f;
}

// ---------------- attention: f32 WMMA scores + softmax + f32 WMMA context ------
__device__ __forceinline__ v8f wmma_f32(v2f a, v2f b, v8f c){
  return __builtin_amdgcn_wmma_f32_16x16x4_f32(false, a, false, b, (short)0, c, false, false);
}

__global__ __launch_bounds__(256) void eqatt_attn(float* ws, int layer)
{
  __shared__ float s_pbv[256];
  s_pbv[threadIdx.x] = ws[WS_PBV + layer*256 + threadIdx.x];
  __syncthreads();

  int wg   = blockIdx.x;             // 512 WGs: 2 per block
  int bb   = wg >> 1;
  int wave = threadIdx.x >> 5;
  int lane = threadIdx.x & 31;
  int lo   = lane & 15;
  int hi   = lane >> 4;
  int q0   = ((wg & 1)*8 + wave) * 16;   // 16-query strip of this wave

  float e4 = ws[WS_ESC + layer];
  float e6 = ws[WS_ESC + 2 + layer];
  const float* gr = ws + WS_GR + bb*(256*24);

  // preload query-side B operands (K = 24 features, 6 wmma steps)
  v2f bq[6];
  #pragma unroll
  for (int s=0;s<6;++s){
    const float* p = gr + (q0 + lo)*24 + 4*s + hi*2;
    bq[s].x = p[0]; bq[s].y = p[1];
  }

  // transposed score strip: 16 key tiles (M = key, N = query)
  v8f st[16];
  #pragma unroll
  for (int kt=0;kt<16;++kt){
    v8f acc = {0.f,0.f,0.f,0.f,0.f,0.f,0.f,0.f};
    #pragma unroll
    for (int s=0;s<6;++s){
      int c = 4*s + hi*2;
      const float* p = gr + (kt*16 + lo)*24 + c;
      v2f a;
      a.x = p[0] * ((c   < 9) ? e4 : e6);
      a.y = p[1] * ((c+1 < 9) ? e4 : e6);
      acc = wmma_f32(a, bq[s], acc);
    }
    #pragma unroll
    for (int g=0; g<8; ++g) acc[g] += s_pbv[kt*16 + g + hi*8];  // key-side bias
    st[kt] = acc;
  }

  // softmax over keys: lane-local reduce + half-wave combine
  float mx = -3.0e38f;
  #pragma unroll
  for (int kt=0;kt<16;++kt)
    #pragma unroll
    for (int g=0;g<8;++g) mx = fmaxf(mx, st[kt][g]);
  mx = fmaxf(mx, __shfl_xor(mx, 16, 32));

  float sum = 0.f;
  #pragma unroll
  for (int kt=0;kt<16;++kt)
    #pragma unroll
    for (int g=0;g<8;++g){ float e = __expf(st[kt][g]-mx); st[kt][g]=e; sum += e; }
  sum += __shfl_xor(sum, 16, 32);
  float rs = 1.f/sum;

  // ctx^T = V^T (80x256) x E (256x16); E built from score regs via half-wave bcasts
  const float* vb = ws + WS_V + bb*(256*80);
  v8f zero8 = {0.f,0.f,0.f,0.f,0.f,0.f,0.f,0.f};
  v8f cacc[5];
  #pragma unroll
  for (int vt=0;vt<5;++vt) cacc[vt] = zero8;

  #pragma unroll
  for (int kt=0;kt<16;++kt){
    #pragma unroll
    for (int s=0;s<4;++s){
      int srcl = lo + ((s >= 2) ? 16 : 0);
      float xl = __shfl(st[kt][(4*s)   & 7], srcl, 32);
      float xh = __shfl(st[kt][(4*s+2) & 7], srcl, 32);
      float yl = __shfl(st[kt][(4*s+1) & 7], srcl, 32);
      float yh = __shfl(st[kt][(4*s+3) & 7], srcl, 32);
      v2f bm; bm.x = hi ? xh : xl; bm.y = hi ? yh : yl;
      int krow = kt*16 + 4*s + hi*2;
      const float* pa = vb + krow*80 + lo;
      #pragma unroll
      for (int vt=0;vt<5;++vt){
        v2f a; a.x = pa[vt*16]; a.y = pa[vt*16 + 80];
        cacc[vt] = wmma_f32(a, bm, cacc[vt]);
      }
    }
  }

  // normalize and scatter ctx[token][v]
  float* cb = ws + WS_CTX + (bb*256 + q0 + lo)*80;
  #pragma unroll
  for (int vt=0;vt<5;++vt)
    #pragma unroll
    for (int g=0;g<8;++g)
      cb[vt*16 + g + hi*8] = cacc[vt][g]*rs;
}

// ---------------- output: 8 Wigner tensor contractions + residual --------------
__device__ __forceinline__ void tp_accum(const float* f, int ni, const float* Q, int nj,
                                         const float* W, int nk, float* out)
{
  for (int i=0;i<ni;++i){
    float fi = f[i];
    for (int j=0;j<nj;++j){
      float c = fi*Q[j];
      const float* w = W + (i*nj+j)*nk;
      for (int k=0;k<nk;++k) out[k] += c*w[k];
    }
  }
}

__global__ __launch_bounds__(256) void eqatt_output(float* ws, int layer)
{
  __shared__ float sW[10648];                // W3J tables 6..13
  for (int i=threadIdx.x; i<10648; i+=256) sW[i] = ws[2670 + i];
  __syncthreads();

  int idx = blockIdx.x*256 + threadIdx.x;
  int b = idx >> 14, pix = idx & 16383;
  int row = pix >> 7, col = pix & 127;
  int bb = b*64 + (row>>4)*8 + (col>>4);
  int n  = (row & 15)*16 + (col & 15);

  float* ft = ws + WS_FEAT + idx*22;
  float f4[9], f6[13];
  for (int i=0;i<9;++i)  f4[i] = ft[i];
  for (int i=0;i<13;++i) f6[i] = ft[9+i];
  const float* cx = ws + WS_CTX + (bb*256+n)*80;
  const float* al = ws + WS_ALPHA + layer*24;

  float d4[9], d6[13], Q[13];
  for (int k=0;k<9;++k)  d4[k]=0.f;
  for (int k=0;k<13;++k) d6[k]=0.f;

  for (int j=0;j<9;++j)  Q[j] = al[0] *cx[j]    + al[1] *cx[9+j]  + al[2] *cx[18+j];
  tp_accum(f4,9, Q,9,  sW+0,    9, d4);
  for (int j=0;j<13;++j) Q[j] = al[3] *cx[27+j] + al[4] *cx[40+j] + al[5] *cx[53+j];
  tp_accum(f4,9, Q,13, sW+729,  9, d4);
  for (int j=0;j<9;++j)  Q[j] = al[6] *cx[j]    + al[7] *cx[9+j]  + al[8] *cx[18+j];
  tp_accum(f6,13,Q,9,  sW+1782, 9, d4);
  for (int j=0;j<13;++j) Q[j] = al[9] *cx[27+j] + al[10]*cx[40+j] + al[11]*cx[53+j];
  tp_accum(f6,13,Q,13, sW+2835, 9, d4);
  for (int j=0;j<9;++j)  Q[j] = al[12]*cx[j]    + al[13]*cx[9+j]  + al[14]*cx[18+j];
  tp_accum(f4,9, Q,9,  sW+4356, 13, d6);
  for (int j=0;j<13;++j) Q[j] = al[15]*cx[27+j] + al[16]*cx[40+j] + al[17]*cx[53+j];
  tp_accum(f4,9, Q,13, sW+5409, 13, d6);
  for (int j=0;j<9;++j)  Q[j] = al[18]*cx[j]    + al[19]*cx[9+j]  + al[20]*cx[18+j];
  tp_accum(f6,13,Q,9,  sW+6930, 13, d6);
  for (int j=0;j<13;++j) Q[j] = al[21]*cx[27+j] + al[22]*cx[40+j] + al[23]*cx[53+j];
  tp_accum(f6,13,Q,13, sW+8451, 13, d6);

  for (int k=0;k<9;++k)  ft[k]   += d4[k];
  for (int k=0;k<13;++k) ft[9+k] += d6[k];
}

// ---------------- 4x kron upsample ----------------
__global__ __launch_bounds__(256) void eqatt_upsample(const float* K4, const float* K6,
                                                      const float* ws, float* out)
{
  int idx = blockIdx.x*256 + threadIdx.x;        // 65536 * 16
  int de = idx & 15; int e = de & 3, d = de >> 2;
  int pix = idx >> 4;
  int b = pix >> 14; int rc = pix & 16383;
  int row = rc >> 7, col = rc & 127;
  const float* f = ws + WS_FEAT + pix*22;
  float k4 = K4[d*4+e], k6 = K6[d*4+e];
  long p = (long)b*262144 + (long)(row*4+d)*512 + (col*4+e);
  float* o4 = out + p*9;
  float* o6 = out + (long)4*262144*9 + p*13;
  for (int c=0;c<9;++c)  o4[c] = f[c]  *k4;
  for (int c=0;c<13;++c) o6[c] = f[9+c]*k6;
}

// ---------------- launch ----------------
extern "C" void kernel_launch(void* const* d_in, const int* in_sizes, int n_in,
                              void* d_out, int out_size, void* d_ws, size_t ws_size,
                              hipStream_t stream)
{
  const float* f4  = (const float*)d_in[0];
  const float* f6  = (const float*)d_in[1];
  const float* ls4 = (const float*)d_in[2];
  const float* ls6 = (const float*)d_in[3];
  const float* pw  = (const float*)d_in[4];
  const float* pb  = (const float*)d_in[5];
  const float* wi4 = (const float*)d_in[6];
  const float* wi6 = (const float*)d_in[7];
  const float* wv  = (const float*)d_in[8];
  const float* wo  = (const float*)d_in[9];
  const float* w4o = (const float*)d_in[10];
  const float* w6o = (const float*)d_in[11];
  const float* K4  = (const float*)d_in[12];
  const float* K6  = (const float*)d_in[13];
  float* out = (float*)d_out;
  float* ws  = (float*)d_ws;
  (void)in_sizes; (void)n_in; (void)out_size; (void)ws_size;

  eqatt_init<<<1, 256, 0, stream>>>(ls4, ls6, pw, pb, wi4, wi6, wv, wo, w4o, w6o, ws);
  eqatt_pack<<<256, 256, 0, stream>>>(f4, f6, ws);
  for (int layer = 0; layer < 2; ++layer){
    eqatt_values<<<256, 256, 0, stream>>>(ws);
    eqatt_attn<<<512, 256, 0, stream>>>(ws, layer);
    eqatt_output<<<256, 256, 0, stream>>>(ws, layer);
  }
  eqatt_upsample<<<4096, 256, 0, stream>>>(K4, K6, ws, out);
}